// DecoderRNN_22617297780738
// MI455X (gfx1250) — compile-verified
//
#include <hip/hip_runtime.h>
#include <math.h>

#define H    1024
#define S    4096
#define V    50257
#define K2H  2048

typedef float v2f __attribute__((ext_vector_type(2)));
typedef float v8f __attribute__((ext_vector_type(8)));

// ---- workspace layout (float offsets) ----
#define OFF_H       0        // h_new (1024)   -- ws[0..2047] is y = cat(h_new, context)
#define OFF_CTX     1024     // context (1024)
#define OFF_U       2048     // folded attention vector u (1024)
#define OFF_SCORES  3072     // scores (4096)
#define OFF_WEIGHTS 7168     // softmax weights (4096)
#define OFF_LOGITS  11264    // logits (50257)
#define OFF_RED     61536    // 50 (max,sum) pairs
#define OFF_MS      61664    // final (M, logSum)
#define NB_LS       50       // log-softmax partial blocks

// ------------------------------------------------------------------
// K1: GRU step. One wave per output j (1024 waves). 37.7 MB stream.
// ------------------------------------------------------------------
__global__ void k_gru(const int* __restrict__ tok,
                      const float* __restrict__ last_ctx,
                      const float* __restrict__ last_h,
                      const float* __restrict__ emb,
                      const float* __restrict__ w_ih,
                      const float* __restrict__ w_hh,
                      const float* __restrict__ b_ih,
                      const float* __restrict__ b_hh,
                      float* __restrict__ ws,
                      float* __restrict__ out_hidden)
{
    const int lane = threadIdx.x & 31;
    const int j = (blockIdx.x * blockDim.x + threadIdx.x) >> 5;
    if (j >= H) return;
    const float* e = emb + (size_t)tok[0] * H;

    float acc[6];
    #pragma unroll
    for (int r = 0; r < 3; ++r) {                 // gi rows j, j+H, j+2H vs x=[e|ctx]
        const float* row = w_ih + (size_t)(j + r * H) * K2H;
        float a = 0.f;
        #pragma unroll
        for (int k = lane * 4; k < K2H; k += 128) {
            const float4 wv = *(const float4*)(row + k);
            const float* xs = (k < H) ? (e + k) : (last_ctx + (k - H));
            const float4 xv = *(const float4*)xs;
            a += wv.x * xv.x + wv.y * xv.y + wv.z * xv.z + wv.w * xv.w;
        }
        acc[r] = a;
    }
    #pragma unroll
    for (int r = 0; r < 3; ++r) {                 // gh rows vs h
        const float* row = w_hh + (size_t)(j + r * H) * H;
        float a = 0.f;
        #pragma unroll
        for (int k = lane * 4; k < H; k += 128) {
            const float4 wv = *(const float4*)(row + k);
            const float4 hv = *(const float4*)(last_h + k);
            a += wv.x * hv.x + wv.y * hv.y + wv.z * hv.z + wv.w * hv.w;
        }
        acc[3 + r] = a;
    }
    #pragma unroll
    for (int r = 0; r < 6; ++r)
        #pragma unroll
        for (int o = 16; o; o >>= 1)
            acc[r] += __shfl_down(acc[r], o, 32);

    if (lane == 0) {
        const float gr = acc[0] + b_ih[j]         + acc[3] + b_hh[j];
        const float gz = acc[1] + b_ih[j + H]     + acc[4] + b_hh[j + H];
        const float gn = acc[2] + b_ih[j + 2 * H];
        const float hn = acc[5] + b_hh[j + 2 * H];
        const float rg = 1.f / (1.f + __expf(-gr));
        const float zg = 1.f / (1.f + __expf(-gz));
        const float nn = tanhf(gn + rg * hn);
        const float hv = (1.f - zg) * nn + zg * last_h[j];
        ws[OFF_H + j] = hv;
        out_hidden[j] = hv;
    }
}

// ------------------------------------------------------------------
// K2: attention fold u[n] = sum_j w_score[j] * atten_W[j, H+n]
//     via V_WMMA_F32_16X16X4_F32. One wave per 16 output columns.
//     A = w_score chunk broadcast into all 16 M-rows (every D row valid),
//     B rows are 16 contiguous floats of atten_W -> lane-contiguous loads.
// ------------------------------------------------------------------
__global__ void k_attn_fold_wmma(const float* __restrict__ atten_W,
                                 const float* __restrict__ w_score,
                                 float* __restrict__ ws)
{
    const int lane = threadIdx.x;          // block = exactly one wave (32)
    const int n0 = blockIdx.x * 16;        // 64 blocks cover H outputs
    const int kh = (lane >> 4) << 1;       // lanes 0-15 -> K={0,1}; 16-31 -> K={2,3}
    const int nl = lane & 15;

    v8f c = {};
    for (int j = 0; j < H; j += 4) {
        v2f a, b;
        a[0] = w_score[j + kh];
        a[1] = w_score[j + kh + 1];
        b[0] = atten_W[(size_t)(j + kh)     * K2H + H + n0 + nl];
        b[1] = atten_W[(size_t)(j + kh + 1) * K2H + H + n0 + nl];
        c = __builtin_amdgcn_wmma_f32_16x16x4_f32(
                false, a, false, b, (short)0, c, false, false);
    }
    if (lane < 16) ws[OFF_U + n0 + lane] = c[0];   // D row M=0
}

// ------------------------------------------------------------------
// K3: scores[s] = u . enc[s,:]. One wave per s (4096 waves, 16.8 MB).
// ------------------------------------------------------------------
__global__ void k_scores(const float* __restrict__ enc,
                         const float* __restrict__ ws_in,
                         float* __restrict__ ws_out)
{
    const int lane = threadIdx.x & 31;
    const int s = (blockIdx.x * blockDim.x + threadIdx.x) >> 5;
    const float* row = enc + (size_t)s * H;
    const float* u = ws_in + OFF_U;
    float a = 0.f;
    #pragma unroll
    for (int k = lane * 4; k < H; k += 128) {
        const float4 ev = *(const float4*)(row + k);
        const float4 uv = *(const float4*)(u + k);
        a += ev.x * uv.x + ev.y * uv.y + ev.z * uv.z + ev.w * uv.w;
    }
    #pragma unroll
    for (int o = 16; o; o >>= 1) a += __shfl_down(a, o, 32);
    if (lane == 0) ws_out[OFF_SCORES + s] = a;
}

// ------------------------------------------------------------------
// K4: softmax over S=4096 scores. Single block, 1024 threads, 4/thread.
// ------------------------------------------------------------------
__global__ void k_softmax_s(float* __restrict__ ws, float* __restrict__ wt_out)
{
    __shared__ float red[32];
    __shared__ float bM, bS;
    const int tid = threadIdx.x, lane = tid & 31, wid = tid >> 5;
    const float* sc = ws + OFF_SCORES;

    float x0 = sc[tid], x1 = sc[tid + 1024], x2 = sc[tid + 2048], x3 = sc[tid + 3072];
    float m = fmaxf(fmaxf(x0, x1), fmaxf(x2, x3));
    #pragma unroll
    for (int o = 16; o; o >>= 1) m = fmaxf(m, __shfl_xor(m, o, 32));
    if (lane == 0) red[wid] = m;
    __syncthreads();
    if (tid < 32) {
        float v = red[tid];
        #pragma unroll
        for (int o = 16; o; o >>= 1) v = fmaxf(v, __shfl_xor(v, o, 32));
        if (tid == 0) bM = v;
    }
    __syncthreads();
    const float M = bM;
    const float e0 = __expf(x0 - M), e1 = __expf(x1 - M),
                e2 = __expf(x2 - M), e3 = __expf(x3 - M);
    float s = e0 + e1 + e2 + e3;
    #pragma unroll
    for (int o = 16; o; o >>= 1) s += __shfl_xor(s, o, 32);
    if (lane == 0) red[wid] = s;
    __syncthreads();
    if (tid < 32) {
        float v = red[tid];
        #pragma unroll
        for (int o = 16; o; o >>= 1) v += __shfl_xor(v, o, 32);
        if (tid == 0) bS = v;
    }
    __syncthreads();
    const float inv = 1.f / bS;
    float* wt = ws + OFF_WEIGHTS;
    wt[tid] = e0 * inv;        wt_out[tid] = e0 * inv;
    wt[tid + 1024] = e1 * inv; wt_out[tid + 1024] = e1 * inv;
    wt[tid + 2048] = e2 * inv; wt_out[tid + 2048] = e2 * inv;
    wt[tid + 3072] = e3 * inv; wt_out[tid + 3072] = e3 * inv;
}

// ------------------------------------------------------------------
// K5: context[k] = sum_s w[s]*enc[s,k]. Column reduce (L2-resident pass).
// ------------------------------------------------------------------
__global__ void k_context(const float* __restrict__ enc,
                          float* __restrict__ ws,
                          float* __restrict__ out_ctx)
{
    const int k = blockIdx.x * blockDim.x + threadIdx.x;   // 0..1023
    const float* wt = ws + OFF_WEIGHTS;
    float a0 = 0.f, a1 = 0.f, a2 = 0.f, a3 = 0.f;
    for (int s = 0; s < S; s += 4) {
        a0 += wt[s]     * enc[(size_t)(s)     * H + k];
        a1 += wt[s + 1] * enc[(size_t)(s + 1) * H + k];
        a2 += wt[s + 2] * enc[(size_t)(s + 2) * H + k];
        a3 += wt[s + 3] * enc[(size_t)(s + 3) * H + k];
    }
    const float a = (a0 + a1) + (a2 + a3);
    ws[OFF_CTX + k] = a;
    out_ctx[k] = a;
}

// ------------------------------------------------------------------
// K6: logits[v] = out_W[v,:] . y + out_b[v].  THE bandwidth kernel:
//     411.5 MB streamed once, wave per row, float4 coalesced.
// ------------------------------------------------------------------
__global__ void k_logits(const float* __restrict__ out_W,
                         const float* __restrict__ out_b,
                         float* __restrict__ ws)
{
    const int lane = threadIdx.x & 31;
    const int v = (blockIdx.x * blockDim.x + threadIdx.x) >> 5;
    if (v >= V) return;
    const float* row = out_W + (size_t)v * K2H;
    const float* y = ws;                        // [h_new | context]
    float a = 0.f;
    #pragma unroll
    for (int k = lane * 4; k < K2H; k += 128) {
        const float4 wv = *(const float4*)(row + k);
        const float4 yv = *(const float4*)(y + k);
        a += wv.x * yv.x + wv.y * yv.y + wv.z * yv.z + wv.w * yv.w;
    }
    #pragma unroll
    for (int o = 16; o; o >>= 1) a += __shfl_down(a, o, 32);
    if (lane == 0) ws[OFF_LOGITS + v] = a + out_b[v];
}

// ------------------------------------------------------------------
// K7a/b/c: log-softmax over V (L2-resident, split-reduce).
// ------------------------------------------------------------------
__global__ void k_ls_part(float* __restrict__ ws)
{
    __shared__ float red[8];
    __shared__ float bM;
    const int tid = threadIdx.x, lane = tid & 31, wid = tid >> 5;
    const float* lg = ws + OFF_LOGITS;
    const int base = blockIdx.x * 1024 + tid;

    float x[4]; int idx[4];
    float m = -1e30f;
    #pragma unroll
    for (int i = 0; i < 4; ++i) {
        idx[i] = base + i * 256;
        x[i] = (idx[i] < V) ? lg[idx[i]] : -1e30f;
        m = fmaxf(m, x[i]);
    }
    #pragma unroll
    for (int o = 16; o; o >>= 1) m = fmaxf(m, __shfl_xor(m, o, 32));
    if (lane == 0) red[wid] = m;
    __syncthreads();
    if (tid < 8) {
        float v = red[tid];
        #pragma unroll
        for (int o = 4; o; o >>= 1) v = fmaxf(v, __shfl_xor(v, o, 32));
        if (tid == 0) bM = v;
    }
    __syncthreads();
    const float M = bM;
    float s = 0.f;
    #pragma unroll
    for (int i = 0; i < 4; ++i)
        if (idx[i] < V) s += __expf(x[i] - M);
    #pragma unroll
    for (int o = 16; o; o >>= 1) s += __shfl_xor(s, o, 32);
    if (lane == 0) red[wid] = s;
    __syncthreads();
    if (tid == 0) {
        float t = 0.f;
        #pragma unroll
        for (int i = 0; i < 8; ++i) t += red[i];
        ws[OFF_RED + 2 * blockIdx.x]     = M;
        ws[OFF_RED + 2 * blockIdx.x + 1] = t;
    }
}

__global__ void k_ls_merge(float* __restrict__ ws)
{
    if (threadIdx.x == 0) {
        float M = -1e30f;
        for (int i = 0; i < NB_LS; ++i) M = fmaxf(M, ws[OFF_RED + 2 * i]);
        float s = 0.f;
        for (int i = 0; i < NB_LS; ++i)
            s += ws[OFF_RED + 2 * i + 1] * __expf(ws[OFF_RED + 2 * i] - M);
        ws[OFF_MS]     = M;
        ws[OFF_MS + 1] = logf(s);
    }
}

__global__ void k_ls_final(const float* __restrict__ ws, float* __restrict__ out)
{
    const int v = blockIdx.x * blockDim.x + threadIdx.x;
    if (v < V) out[v] = ws[OFF_LOGITS + v] - ws[OFF_MS] - ws[OFF_MS + 1];
}

// ------------------------------------------------------------------
extern "C" void kernel_launch(void* const* d_in, const int* in_sizes, int n_in,
                              void* d_out, int out_size, void* d_ws, size_t ws_size,
                              hipStream_t stream)
{
    (void)in_sizes; (void)n_in; (void)out_size; (void)ws_size;
    const int*   tok      = (const int*)  d_in[0];
    const float* last_ctx = (const float*)d_in[1];
    const float* last_h   = (const float*)d_in[2];
    const float* enc      = (const float*)d_in[3];   // (S,1,H) flat
    const float* emb      = (const float*)d_in[4];
    const float* w_ih     = (const float*)d_in[5];
    const float* w_hh     = (const float*)d_in[6];
    const float* b_ih     = (const float*)d_in[7];
    const float* b_hh     = (const float*)d_in[8];
    const float* atten_W  = (const float*)d_in[9];
    // d_in[10] = atten_b: provably cancels inside softmax -> unused
    const float* w_score  = (const float*)d_in[11];
    const float* out_W    = (const float*)d_in[12];
    const float* out_b    = (const float*)d_in[13];

    float* ws  = (float*)d_ws;
    float* out = (float*)d_out;
    float* out_logp    = out;               // (V)
    float* out_ctx     = out + V;           // (H)
    float* out_hidden  = out + V + H;       // (H)
    float* out_weights = out + V + 2 * H;   // (S)

    // K1: GRU -> h_new (1024 waves)
    k_gru<<<128, 256, 0, stream>>>(tok, last_ctx, last_h, emb, w_ih, w_hh,
                                   b_ih, b_hh, ws, out_hidden);
    // K2: WMMA attention fold u = A2^T w_score (64 single-wave blocks)
    k_attn_fold_wmma<<<64, 32, 0, stream>>>(atten_W, w_score, ws);
    // K3: scores (4096 waves)
    k_scores<<<512, 256, 0, stream>>>(enc, ws, ws);
    // K4: softmax over S
    k_softmax_s<<<1, 1024, 0, stream>>>(ws, out_weights);
    // K5: context
    k_context<<<H / 256, 256, 0, stream>>>(enc, ws, out_ctx);
    // K6: vocab projection (dominant 411.5 MB stream)
    k_logits<<<(V + 7) / 8, 256, 0, stream>>>(out_W, out_b, ws);
    // K7: log-softmax over V
    k_ls_part <<<NB_LS, 256, 0, stream>>>(ws);
    k_ls_merge<<<1, 64, 0, stream>>>(ws);
    k_ls_final<<<(V + 255) / 256, 256, 0, stream>>>(ws, out_logp);
}